// ActionableRGM_21431886807534
// MI455X (gfx1250) — compile-verified
//
#include <hip/hip_runtime.h>
#include <hip/hip_bf16.h>
#include <math.h>

// Problem constants (from reference): B=4, L=512, D=257, M=128
#define BB 4
#define LL 512
#define DD 257
#define MM 128
#define DA (DD + 1)      // augmented matrix width for the solve
#define DP 272           // 257 padded up to 17*16 for WMMA tiling
#define BL (BB * LL)     // 2048 rows in the batched GEMM

typedef __attribute__((ext_vector_type(2))) float v2f;
typedef __attribute__((ext_vector_type(8))) float v8f;

// ---------------------------------------------------------------------------
// Kernel 1: one-workgroup Gaussian elimination with partial pivoting.
// Solves S * w0 = z0 (257x257). Augmented matrix lives in global workspace,
// w0 kept in LDS during back-substitution.
// ---------------------------------------------------------------------------
__global__ void rgm_ge_solve(const float* __restrict__ S,
                             const float* __restrict__ z0,
                             float* __restrict__ A,      // DD x DA workspace
                             float* __restrict__ w0g) {  // DD
    const int nt = 256;
    const int tid = threadIdx.x;
    __shared__ float red[256];
    __shared__ float w[DD];
    __shared__ int pivIdx;

    // Build augmented [S | z0]
    for (int idx = tid; idx < DD * DA; idx += nt) {
        int r = idx / DA, c = idx % DA;
        A[idx] = (c < DD) ? S[r * DD + c] : z0[r];
    }
    __threadfence_block();
    __syncthreads();

    // Forward elimination with partial pivoting
    for (int k = 0; k < DD; ++k) {
        if (tid == 0) {
            int p = k;
            float best = fabsf(A[k * DA + k]);
            for (int r = k + 1; r < DD; ++r) {
                float v = fabsf(A[r * DA + k]);
                if (v > best) { best = v; p = r; }
            }
            pivIdx = p;
        }
        __syncthreads();
        int p = pivIdx;
        if (p != k) {
            for (int c = k + tid; c < DA; c += nt) {
                float t = A[k * DA + c];
                A[k * DA + c] = A[p * DA + c];
                A[p * DA + c] = t;
            }
            __threadfence_block();
        }
        __syncthreads();
        float pivval = A[k * DA + k];
        for (int r = k + 1 + tid; r < DD; r += nt) {
            float f = A[r * DA + k] / pivval;
            for (int c = k; c < DA; ++c) A[r * DA + c] -= f * A[k * DA + c];
        }
        __threadfence_block();
        __syncthreads();
    }

    // Back substitution (parallel dot products, serial over k)
    for (int k = DD - 1; k >= 0; --k) {
        float part = 0.0f;
        for (int c = k + 1 + tid; c < DD; c += nt) part += A[k * DA + c] * w[c];
        red[tid] = part;
        __syncthreads();
        for (int s = 128; s > 0; s >>= 1) {
            if (tid < s) red[tid] += red[tid + s];
            __syncthreads();
        }
        if (tid == 0) w[k] = (A[k * DA + DD] - red[0]) / A[k * DA + k];
        __syncthreads();
    }
    for (int c = tid; c < DD; c += nt) w0g[c] = w[c];
}

// ---------------------------------------------------------------------------
// Kernel 2: zero-pad S (257x257) -> Sp (272x272)
// ---------------------------------------------------------------------------
__global__ void rgm_pad_S(const float* __restrict__ S, float* __restrict__ Sp) {
    int idx = blockIdx.x * blockDim.x + threadIdx.x;
    if (idx < DP * DP) {
        int i = idx / DP, j = idx % DP;
        Sp[idx] = (i < DD && j < DD) ? S[i * DD + j] : 0.0f;
    }
}

// ---------------------------------------------------------------------------
// Kernel 3: the scan in irrep space. One block per batch, one thread per
// Givens pair; the (w1,w2) state stays in registers across all 512 steps.
// Writes Wp as (B*L) x 272, zero-padded columns.
// ---------------------------------------------------------------------------
__global__ void rgm_recurrence(const float* __restrict__ x,   // B,L,2
                               const float* __restrict__ om,  // M,2
                               const float* __restrict__ w0,  // D
                               float* __restrict__ Wp) {      // (B*L) x DP
    int b = blockIdx.x;
    int m = threadIdx.x;  // 0..127
    float o0 = om[2 * m + 0];
    float o1 = om[2 * m + 1];
    float w1 = w0[2 * m + 1];
    float w2 = w0[2 * m + 2];
    float w00 = w0[0];
    const float* xb = x + (size_t)b * LL * 2;
    float* Wb = Wp + (size_t)b * LL * DP;
    for (int t = 0; t < LL; ++t) {
        float ang = xb[2 * t] * o0 + xb[2 * t + 1] * o1;
        float c, s;
        sincosf(ang, &s, &c);
        float nw1 = c * w1 - s * w2;
        float nw2 = s * w1 + c * w2;
        w1 = nw1; w2 = nw2;
        Wb[t * DP + 2 * m + 1] = w1;
        Wb[t * DP + 2 * m + 2] = w2;
        if (m == 0) Wb[t * DP + 0] = w00;           // component 0 is invariant
        if (m < DP - DD) Wb[t * DP + DD + m] = 0.0f; // pad cols 257..271
    }
}

// ---------------------------------------------------------------------------
// Kernel 4: outputs = Wp * Sp^T via V_WMMA_F32_16X16X4_F32.
// One wave (32 threads) per 16x16 output tile; K loop of 68 wmma steps.
// A frag (16x4 f32): lane<16 -> row=lane, K={k,k+1}; lane>=16 -> K={k+2,k+3}.
// B frag (4x16):     lane<16 -> col=lane, K={k,k+1}; lane>=16 -> K={k+2,k+3}.
//   B[k][n] = Sp[n][k]  (since C = W * S^T).
// C/D frag: VGPR v -> row mT*16+v+8*half, col nT*16+(lane&15).
// ---------------------------------------------------------------------------
__global__ void rgm_gemm_wmma(const float* __restrict__ Wp,  // (B*L) x DP
                              const float* __restrict__ Sp,  // DP x DP
                              float* __restrict__ out) {     // (B*L) x DD
    int tile = blockIdx.x;
    int mT = tile / (DP / 16);
    int nT = tile % (DP / 16);
    int lane = threadIdx.x;
    int half = lane >> 4;
    int l15 = lane & 15;

    const float* Arow = Wp + (size_t)(mT * 16 + l15) * DP;
    const float* Brow = Sp + (size_t)(nT * 16 + l15) * DP;

    v8f acc = {};
    for (int k = 0; k < DP; k += 4) {
        int kk = k + 2 * half;
        v2f a; a.x = Arow[kk]; a.y = Arow[kk + 1];
        v2f b; b.x = Brow[kk]; b.y = Brow[kk + 1];
        acc = __builtin_amdgcn_wmma_f32_16x16x4_f32(
            /*neg_a=*/false, a, /*neg_b=*/false, b,
            /*c_mod=*/(short)0, acc, /*reuse_a=*/false, /*reuse_b=*/false);
    }

    int col = nT * 16 + l15;
    if (col < DD) {
        #pragma unroll
        for (int v = 0; v < 8; ++v) {
            int row = mT * 16 + v + 8 * half;
            out[(size_t)row * DD + col] = acc[v];
        }
    }
}

// ---------------------------------------------------------------------------
// Kernel 5: z_final = outputs[:, L-1, :]
// ---------------------------------------------------------------------------
__global__ void rgm_final(const float* __restrict__ outs, float* __restrict__ zf) {
    int i = blockIdx.x * blockDim.x + threadIdx.x;
    if (i < BB * DD) {
        int b = i / DD, d = i % DD;
        zf[i] = outs[((size_t)b * LL + (LL - 1)) * DD + d];
    }
}

extern "C" void kernel_launch(void* const* d_in, const int* in_sizes, int n_in,
                              void* d_out, int out_size, void* d_ws, size_t ws_size,
                              hipStream_t stream) {
    const float* x  = (const float*)d_in[0];  // (4,512,2)
    const float* z0 = (const float*)d_in[1];  // (257,)
    const float* om = (const float*)d_in[2];  // (128,2)
    const float* S  = (const float*)d_in[3];  // (257,257)

    float* outs = (float*)d_out;              // (4,512,257) flat
    float* zfin = outs + (size_t)BB * LL * DD;

    // Workspace layout (floats): w0 | GE augmented | padded S | padded W
    float* ws   = (float*)d_ws;
    float* w0   = ws;                          // 257   (pad to 320)
    float* geA  = ws + 320;                    // 257*258 = 66306 (pad to 66560)
    float* Sp   = ws + 320 + 66560;            // 272*272 = 73984
    float* Wp   = Sp + DP * DP;                // 2048*272 = 557056
    // total ~2.8 MB

    rgm_ge_solve<<<1, 256, 0, stream>>>(S, z0, geA, w0);
    rgm_pad_S<<<(DP * DP + 255) / 256, 256, 0, stream>>>(S, Sp);
    rgm_recurrence<<<BB, MM, 0, stream>>>(x, om, w0, Wp);
    rgm_gemm_wmma<<<(BL / 16) * (DP / 16), 32, 0, stream>>>(Wp, Sp, outs);
    rgm_final<<<(BB * DD + 255) / 256, 256, 0, stream>>>(outs, zfin);
}